// FocalLoss_41970420416787
// MI455X (gfx1250) — compile-verified
//
#include <hip/hip_runtime.h>

// Problem constants (match reference)
static constexpr int B = 8;
static constexpr int A = 120000;
static constexpr int C = 80;
static constexpr int M = 32;

static constexpr int ABLOCKS = (A + 255) / 256;   // 469 blocks per image for assign kernel
static constexpr int CBLOCKS = 640;               // blocks per image for focal kernel
static constexpr int NV4     = A * C / 4;         // 2,400,000 float4 per image

typedef __attribute__((ext_vector_type(2))) float v2f;
typedef __attribute__((ext_vector_type(4))) float v4f;
typedef __attribute__((ext_vector_type(8))) float v8f;

// ---------------------------------------------------------------------------
// Wave-level sum using V_WMMA_F32_16X16X4_F32.
// A-matrix layout (16x4 f32, wave32): lanes 0-15 -> rows m=0..15, VGPR0=K0,
// VGPR1=K1; lanes 16-31 -> rows m=0..15, VGPR0=K2, VGPR1=K3.
// With A = (v, 0) per lane and B = all ones:  D[m][n] = v[m] + v[m+16].
// C/D layout: d[i] lanes 0-15 = row m=i, lanes 16-31 = row m=i+8, so
// sum_{i<8} d[i] gives half-wave row-sum totals; one xor-16 shuffle finishes.
// EXEC must be all ones at the call site (uniform control flow only).
// ---------------------------------------------------------------------------
__device__ __forceinline__ float wave_sum_wmma(float v) {
    v2f a; a[0] = v;    a[1] = 0.0f;
    v2f b; b[0] = 1.0f; b[1] = 1.0f;
    v8f c = {};
    v8f d = __builtin_amdgcn_wmma_f32_16x16x4_f32(
        /*neg_a=*/false, a, /*neg_b=*/false, b,
        /*c_mod=*/(short)0, c, /*reuse_a=*/false, /*reuse_b=*/false);
    float s = 0.0f;
#pragma unroll
    for (int i = 0; i < 8; ++i) s += d[i];
    s += __shfl_xor(s, 16, 32);
    return s;  // full wave total in every lane
}

// Block (256 threads = 8 waves) sum; result valid on threadIdx.x == 0.
// Deterministic: fixed wave order, fixed 8-term final add.
__device__ __forceinline__ float block_sum_wmma(float v, float* red8) {
    float s = wave_sum_wmma(v);
    const int tid = threadIdx.x;
    if ((tid & 31) == 0) red8[tid >> 5] = s;
    __syncthreads();
    float tot = 0.0f;
    if (tid == 0) {
#pragma unroll
        for (int i = 0; i < 8; ++i) tot += red8[i];
    }
    __syncthreads();  // safe LDS reuse by caller
    return tot;
}

// ---------------------------------------------------------------------------
// Kernel 1: anchor assignment + smooth-L1 regression loss partials.
// Grid: (ABLOCKS, B) x 256 threads. One thread per anchor.
// ---------------------------------------------------------------------------
__global__ void assign_reg_kernel(const float* __restrict__ regressions,
                                  const float* __restrict__ anchors,
                                  const float* __restrict__ annotations,
                                  int*   __restrict__ state,
                                  float* __restrict__ reg_p,
                                  float* __restrict__ pos_p) {
    __shared__ float ann[M * 5];
    __shared__ float red8[8];
    const int b   = blockIdx.y;
    const int tid = threadIdx.x;
    if (tid < M * 5) ann[tid] = annotations[b * M * 5 + tid];
    __syncthreads();

    const int  a      = blockIdx.x * blockDim.x + tid;
    const bool active = (a < A);
    const int  ai     = active ? a : (A - 1);  // clamp: keep control flow uniform

    const float ax1 = anchors[ai * 4 + 0], ay1 = anchors[ai * 4 + 1];
    const float ax2 = anchors[ai * 4 + 2], ay2 = anchors[ai * 4 + 3];
    const float aw  = ax2 - ax1, ah = ay2 - ay1;
    const float acx = ax1 + 0.5f * aw, acy = ay1 + 0.5f * ah;
    const float areaA = aw * ah;

    float best = -2.0f;  // any real iou (>= -1) beats this; ties keep first m
    int   bestm = 0;
#pragma unroll 4
    for (int m = 0; m < M; ++m) {
        const float bx1 = ann[m * 5 + 0], by1 = ann[m * 5 + 1];
        const float bx2 = ann[m * 5 + 2], by2 = ann[m * 5 + 3];
        const float lb  = ann[m * 5 + 4];
        float iw = fminf(ax2, bx2) - fmaxf(ax1, bx1); iw = fmaxf(iw, 0.0f);
        float ih = fminf(ay2, by2) - fmaxf(ay1, by1); ih = fmaxf(ih, 0.0f);
        const float inter = iw * ih;
        const float ua    = fmaxf(areaA + (bx2 - bx1) * (by2 - by1) - inter, 1e-8f);
        float iou = inter / ua;
        if (lb == -1.0f) iou = -1.0f;          // never match padding
        if (iou > best) { best = iou; bestm = m; }
    }
    const bool pos = (best >= 0.5f);
    const bool neg = (best < 0.4f);
    const int  st  = pos ? (int)ann[bestm * 5 + 4] : (neg ? -1 : -2);
    if (active) state[b * A + a] = st;

    float rl = 0.0f;
    if (pos && active) {
        const float bx1 = ann[bestm * 5 + 0], by1 = ann[bestm * 5 + 1];
        const float bx2 = ann[bestm * 5 + 2], by2 = ann[bestm * 5 + 3];
        float gw = bx2 - bx1, gh = by2 - by1;
        const float gcx = bx1 + 0.5f * gw, gcy = by1 + 0.5f * gh;  // ctr from unclamped wh
        gw = fmaxf(gw, 1.0f); gh = fmaxf(gh, 1.0f);
        const float t0 = (gcx - acx) / aw * 10.0f;   // /0.1
        const float t1 = (gcy - acy) / ah * 10.0f;
        const float t2 = __logf(gw / aw) * 5.0f;     // /0.2
        const float t3 = __logf(gh / ah) * 5.0f;
        const float* rg = regressions + ((size_t)b * A + (size_t)a) * 4;
        const float d0 = fabsf(t0 - rg[0]);
        const float d1 = fabsf(t1 - rg[1]);
        const float d2 = fabsf(t2 - rg[2]);
        const float d3 = fabsf(t3 - rg[3]);
        const float th = 1.0f / 9.0f, sub = 0.5f / 9.0f;
        rl  = (d0 <= th) ? 4.5f * d0 * d0 : d0 - sub;
        rl += (d1 <= th) ? 4.5f * d1 * d1 : d1 - sub;
        rl += (d2 <= th) ? 4.5f * d2 * d2 : d2 - sub;
        rl += (d3 <= th) ? 4.5f * d3 * d3 : d3 - sub;
    }
    const float pc = (pos && active) ? 1.0f : 0.0f;

    // Uniform control flow here: all 256 lanes execute the WMMA reductions.
    const float rsum = block_sum_wmma(rl, red8);
    const float psum = block_sum_wmma(pc, red8);
    if (tid == 0) {
        reg_p[b * gridDim.x + blockIdx.x] = rsum;
        pos_p[b * gridDim.x + blockIdx.x] = psum;
    }
}

// ---------------------------------------------------------------------------
// Kernel 2: focal classification loss. Grid: (CBLOCKS, B) x 256 threads.
// Streams [A,C] per image as float4 with NT hint (read-once, 307 MB total).
// One log per element: bce = t ? -log(p) : -log(1-p).
// ---------------------------------------------------------------------------
__global__ void focal_cls_kernel(const float* __restrict__ classifications,
                                 const int*   __restrict__ state,
                                 float* __restrict__ cls_p) {
    __shared__ float red8[8];
    const int b = blockIdx.y;
    const v4f* __restrict__ p = (const v4f*)(classifications + (size_t)b * A * C);
    const int* __restrict__ st_img = state + (size_t)b * A;

    float acc = 0.0f;
    const int stride = gridDim.x * blockDim.x;
    for (int f = blockIdx.x * blockDim.x + threadIdx.x; f < NV4; f += stride) {
        const v4f v = __builtin_nontemporal_load(p + f);
        const int e     = f * 4;        // element index within image (< 9.6M)
        const int a     = e / C;        // C = 80, float4 never crosses an anchor
        const int cbase = e - a * C;
        const int s     = st_img[a];    // class id (>=0), -1 = neg, -2 = ignore
        if (s != -2) {
#pragma unroll
            for (int i = 0; i < 4; ++i) {
                float x = v[i];
                x = fminf(fmaxf(x, 1e-4f), 1.0f - 1e-4f);
                const bool t1 = (s == cbase + i);           // false for all c when s==-1
                const float fw  = t1 ? (1.0f - x) : x;
                const float af  = t1 ? 0.25f : 0.75f;
                const float bce = t1 ? -__logf(x) : -__logf(1.0f - x);
                acc += af * fw * fw * bce;
            }
        }
    }
    // Loop reconverges; uniform here.
    const float bsum = block_sum_wmma(acc, red8);
    if (threadIdx.x == 0) cls_p[b * gridDim.x + blockIdx.x] = bsum;
}

// ---------------------------------------------------------------------------
// Kernel 3: deterministic finalize. One block of 256 threads.
// ---------------------------------------------------------------------------
__global__ void finalize_kernel(const float* __restrict__ annotations,
                                const float* __restrict__ reg_p,
                                const float* __restrict__ pos_p,
                                const float* __restrict__ cls_p,
                                float* __restrict__ out) {
    __shared__ float red8[8];
    __shared__ float res[2];
    const int tid = threadIdx.x;
    if (tid == 0) { res[0] = 0.0f; res[1] = 0.0f; }
    __syncthreads();

    for (int b = 0; b < B; ++b) {
        float pv = 0.0f, rv = 0.0f, cv = 0.0f;
        for (int i = tid; i < ABLOCKS; i += blockDim.x) {
            pv += pos_p[b * ABLOCKS + i];
            rv += reg_p[b * ABLOCKS + i];
        }
        for (int i = tid; i < CBLOCKS; i += blockDim.x)
            cv += cls_p[b * CBLOCKS + i];

        const float np = block_sum_wmma(pv, red8);
        const float rs = block_sum_wmma(rv, red8);
        const float cs = block_sum_wmma(cv, red8);
        if (tid == 0) {
            int nvalid = 0;
            for (int m = 0; m < M; ++m)
                if (annotations[b * M * 5 + m * 5 + 4] != -1.0f) ++nvalid;
            float closs = cs / fmaxf(np, 1.0f);
            float rloss = (np > 0.5f) ? rs / (fmaxf(np, 1.0f) * 4.0f) : 0.0f;
            if (nvalid == 0) { closs = 0.0f; rloss = 0.0f; }
            res[0] += closs; res[1] += rloss;
        }
        __syncthreads();
    }
    if (tid == 0) {
        out[0] = res[0] / (float)B;   // mean classification loss
        out[1] = res[1] / (float)B;   // mean regression loss
    }
}

// ---------------------------------------------------------------------------
extern "C" void kernel_launch(void* const* d_in, const int* in_sizes, int n_in,
                              void* d_out, int out_size, void* d_ws, size_t ws_size,
                              hipStream_t stream) {
    (void)in_sizes; (void)n_in; (void)out_size; (void)ws_size;
    const float* classifications = (const float*)d_in[0];  // [B,A,C]
    const float* regressions     = (const float*)d_in[1];  // [B,A,4]
    const float* anchors         = (const float*)d_in[2];  // [1,A,4]
    const float* annotations     = (const float*)d_in[3];  // [B,M,5]
    float* out = (float*)d_out;                            // [2]

    // Workspace layout
    int*   state = (int*)d_ws;                                            // B*A ints
    float* reg_p = (float*)((char*)d_ws + (size_t)B * A * sizeof(int));   // B*ABLOCKS
    float* pos_p = reg_p + (size_t)B * ABLOCKS;                           // B*ABLOCKS
    float* cls_p = pos_p + (size_t)B * ABLOCKS;                           // B*CBLOCKS

    dim3 gA(ABLOCKS, B);
    assign_reg_kernel<<<gA, 256, 0, stream>>>(regressions, anchors, annotations,
                                              state, reg_p, pos_p);
    dim3 gB(CBLOCKS, B);
    focal_cls_kernel<<<gB, 256, 0, stream>>>(classifications, state, cls_p);

    finalize_kernel<<<1, 256, 0, stream>>>(annotations, reg_p, pos_p, cls_p, out);
}